// RoPE_8349416423518
// MI455X (gfx1250) — compile-verified
//
#include <hip/hip_runtime.h>

// RoPE forward, MI455X (gfx1250, wave32).
//
// Roofline: 256MB in + 256MB out + 4MB pos ~= 516MB @ 23.3 TB/s -> ~22us.
// Purely elementwise 2x2 rotations with per-(token,k) angles: no shared
// matrix operand exists, so WMMA/SWMMAC cannot express this op -- the win
// is saturating HBM with B128 non-temporal traffic and keeping the VALU
// work under the memory time (inv_freq hoisted; 2 hw sincos per 16B).
//
// Layout: one wave32 per token (D_K=128 floats = 32 lanes x float4), so each
// wave emits exactly one coalesced 512B global_load_b128 and one 512B
// global_store_b128 per token. Grid-stride over tokens with a one-iteration
// prefetch (global_prefetch_b8) to deepen the memory pipeline.
//
// Trans ops via raw AMDGCN builtins (header-independent):
//   v_exp_f32 for inv_freq (hoisted), v_sin_f32/v_cos_f32 for the rotation.
// V_SIN/V_COS take input in revolutions with a limited domain, so we range-
// reduce explicitly: frac = rev - rint(rev) in [-0.5, 0.5].

typedef float v4f __attribute__((ext_vector_type(4)));

__global__ __launch_bounds__(256) void rope_fwd_f32(
    const float* __restrict__ x,
    const long long* __restrict__ tpos,   // int64 positions, values < 4096
    float* __restrict__ out,
    int n_tokens)
{
    const int lane   = threadIdx.x & 31;
    const int wid    = (int)(blockIdx.x * (blockDim.x >> 5) + (threadIdx.x >> 5));
    const int nwaves = (int)((gridDim.x * blockDim.x) >> 5);

    // inv_freq[k] = 10000^(-k/64) = exp2(-k * log2(10000)/64).
    // Depends only on the lane -> hoist the two v_exp_f32 out of the loop.
    // Fold the 1/(2*pi) revolution prescale in here too (it's a constant
    // factor on the angle), saving a multiply in the loop.
    const float LOG2_THETA_OVER_HALF_DK = 0.20762050593046007f; // log2(10000)/64
    const float INV_2PI = 0.15915494309189535f;
    const float k0   = (float)(2 * lane);
    const float rf0  = INV_2PI * __builtin_amdgcn_exp2f(-k0 * LOG2_THETA_OVER_HALF_DK);
    const float rf1  = INV_2PI * __builtin_amdgcn_exp2f(-(k0 + 1.0f) * LOG2_THETA_OVER_HALF_DK);

    for (int t = wid; t < n_tokens; t += nwaves) {
        const long long base = (long long)t * 128 + (lane << 2);

        // Prefetch next iteration's row (gfx1250 global_prefetch_b8).
        if (t + nwaves < n_tokens) {
            __builtin_prefetch(x + ((long long)(t + nwaves) * 128 + (lane << 2)), 0, 0);
        }

        // Streaming data: non-temporal to keep the 192MB L2 clean.
        v4f v = __builtin_nontemporal_load((const v4f*)(x + base));

        // Position is wave-uniform; value < 4096 so an i32 convert is exact.
        const float p = (float)(int)tpos[t];

        // Angle in revolutions, range-reduced to [-0.5, 0.5] for hw sin/cos.
        const float rev0 = p * rf0;
        const float rev1 = p * rf1;
        const float r0   = rev0 - __builtin_rintf(rev0);   // v_rndne_f32
        const float r1   = rev1 - __builtin_rintf(rev1);

        const float s0 = __builtin_amdgcn_sinf(r0);        // v_sin_f32
        const float c0 = __builtin_amdgcn_cosf(r0);        // v_cos_f32
        const float s1 = __builtin_amdgcn_sinf(r1);
        const float c1 = __builtin_amdgcn_cosf(r1);

        // out[2k]   = cos*xe - sin*xo
        // out[2k+1] = sin*xe + cos*xo   (interleaved back in place)
        v4f o;
        o.x = __builtin_fmaf(c0, v.x, -(s0 * v.y));
        o.y = __builtin_fmaf(s0, v.x,   c0 * v.y);
        o.z = __builtin_fmaf(c1, v.z, -(s1 * v.w));
        o.w = __builtin_fmaf(s1, v.z,   c1 * v.w);

        __builtin_nontemporal_store(o, (v4f*)(out + base));
    }
}

extern "C" void kernel_launch(void* const* d_in, const int* in_sizes, int n_in,
                              void* d_out, int out_size, void* d_ws, size_t ws_size,
                              hipStream_t stream)
{
    const float*     x    = (const float*)d_in[0];
    const long long* tpos = (const long long*)d_in[1];   // jnp.int64 per reference
    float*           out  = (float*)d_out;

    const int n_tokens = in_sizes[1];          // 4*32*4096 = 524,288 tokens

    // 8192 blocks x 256 threads = 65,536 waves -> 8 tokens per wave:
    // enough waves to saturate every WGP while keeping a grid-stride loop
    // so the prefetch has a target.
    dim3 grid(8192), block(256);
    rope_fwd_f32<<<grid, block, 0, stream>>>(x, tpos, out, n_tokens);
}